// MHAttention_23862838297135
// MI455X (gfx1250) — compile-verified
//
#include <hip/hip_runtime.h>

typedef float v2f __attribute__((ext_vector_type(2)));
typedef float v8f __attribute__((ext_vector_type(8)));

#define B_  2
#define S_  2048
#define D_  1024
#define H_  16
#define DK_ 64
#define N3D_ (3 * D_)
#define MROWS_ (B_ * S_)          // 4096
#define THETA_ 10000.0f

__device__ __forceinline__ v8f wmma_f32(v2f a, v2f b, v8f c) {
    // V_WMMA_F32_16X16X4_F32 : D = A(16x4,f32) * B(4x16,f32) + C(16x16,f32)
    return __builtin_amdgcn_wmma_f32_16x16x4_f32(
        /*neg_a=*/false, a, /*neg_b=*/false, b,
        /*c_mod=*/(short)0, c, /*reuse_a=*/false, /*reuse_b=*/false);
}

// ---------------------------------------------------------------------------
// Kernel 1: QKV projection + RoPE epilogue.
// X:(4096,1024) @ Wqkv:(1024,3072) -> Q,K,V each (B,H,S,DK) in workspace.
// Block = 256 threads = 8 waves; wave computes one 16x16 tile; block = 32x64.
// ---------------------------------------------------------------------------
__global__ __launch_bounds__(256) void qkv_rope_kernel(
    const float* __restrict__ X, const float* __restrict__ W,
    const int* __restrict__ tokpos,
    float* __restrict__ Qo, float* __restrict__ Ko, float* __restrict__ Vo) {
    const int lane = threadIdx.x & 31;
    const int wave = threadIdx.x >> 5;
    const int l16  = lane & 15;
    const int half = lane >> 4;
    const int rowBase = blockIdx.y * 32 + (wave >> 2) * 16;
    const int colBase = blockIdx.x * 64 + (wave & 3) * 16;

    const float* Arow = X + (size_t)(rowBase + l16) * D_ + 2 * half;  // A[M=l16][k+2*half]
    const float* Bcol = W + (size_t)(2 * half) * N3D_ + colBase + l16; // B[k+2*half][N=l16]

    v8f acc = {};
    #pragma unroll 4
    for (int k = 0; k < D_; k += 4) {
        v2f a; a.x = Arow[k];               a.y = Arow[k + 1];
        v2f b; b.x = Bcol[(size_t)k * N3D_]; b.y = Bcol[(size_t)(k + 1) * N3D_];
        acc = wmma_f32(a, b, acc);
    }

    // Epilogue: C layout -> lane l16 = col, vgpr t = row (t + 8*half).
    #pragma unroll
    for (int t = 0; t < 8; ++t) {
        const int M  = t + 8 * half;
        const int gm = rowBase + M;            // global row in [0, B*S)
        const int gn = colBase + l16;          // global col in [0, 3*D)
        const int s  = gm & (S_ - 1);
        const int b  = gm >> 11;               // / S_
        const int sel = gn >> 10;              // 0=Q 1=K 2=V
        const int d   = gn & (D_ - 1);
        const int h   = d >> 6;
        const int dd  = d & (DK_ - 1);
        float val = acc[t];
        float out;
        if (sel < 2) {                         // RoPE on Q and K
            const float partner = __shfl_xor(val, 1, 32);   // col N xor 1
            const float pos = (float)tokpos[s];
            const float fexp = (float)(dd & ~1) * (1.0f / (float)DK_);
            const float inv_freq = __powf(THETA_, -fexp);
            float sn, cs;
            __sincosf(pos * inv_freq, &sn, &cs);
            out = (dd & 1) ? (partner * sn + val * cs)      // r2 = x1*sin + x2*cos
                           : (val * cs - partner * sn);     // r1 = x1*cos - x2*sin
        } else {
            out = val;
        }
        float* dst = (sel == 0) ? Qo : (sel == 1) ? Ko : Vo;
        dst[(size_t)(((b * H_ + h) * S_) + s) * DK_ + dd] = out;
    }
}

// ---------------------------------------------------------------------------
// Kernel 2: causal flash attention, f32 WMMA. One wave (block of 32) per
// (b,h, 16-query tile). Q,K,V in (B,H,S,DK); output in (B,S,D).
// ---------------------------------------------------------------------------
__global__ __launch_bounds__(32) void flash_kernel(
    const float* __restrict__ Q, const float* __restrict__ K,
    const float* __restrict__ V, float* __restrict__ Oout) {
    __shared__ float pbuf[16][17];             // P-tile transpose staging
    const int lane = threadIdx.x & 31;
    const int l16  = lane & 15;
    const int half = lane >> 4;
    const int qt = blockIdx.x;                 // query tile 0..127
    const int bh = blockIdx.y;                 // b*H + h
    const int b  = bh >> 4, h = bh & 15;
    const int qBase = qt * 16;

    const float* Qp = Q + (size_t)bh * S_ * DK_;
    const float* Kp = K + (size_t)bh * S_ * DK_;
    const float* Vp = V + (size_t)bh * S_ * DK_;

    // Resident Q fragments, pre-scaled by 1/sqrt(DK) = 0.125
    v2f qa[16];
    #pragma unroll
    for (int kk = 0; kk < 16; ++kk) {
        const float* p = Qp + (size_t)(qBase + l16) * DK_ + kk * 4 + 2 * half;
        qa[kk].x = p[0] * 0.125f;
        qa[kk].y = p[1] * 0.125f;
    }

    v8f o0 = {}, o1 = {}, o2 = {}, o3 = {};
    float m_i[8], l_i[8];
    #pragma unroll
    for (int t = 0; t < 8; ++t) { m_i[t] = -1e30f; l_i[t] = 0.0f; }

    for (int kt = 0; kt <= qt; ++kt) {
        const int kBase = kt * 16;

        // S_tile = (Q/8) @ K^T  : 16 WMMA steps over DK=64
        v8f sc = {};
        #pragma unroll
        for (int kk = 0; kk < 16; ++kk) {
            const float* p = Kp + (size_t)(kBase + l16) * DK_ + kk * 4 + 2 * half;
            v2f bf; bf.x = p[0]; bf.y = p[1];
            sc = wmma_f32(qa[kk], bf, sc);
        }

        // causal mask + online softmax (rows live in lane halves)
        float alpha[8];
        #pragma unroll
        for (int t = 0; t < 8; ++t) {
            const int row = qBase + t + 8 * half;
            const int col = kBase + l16;
            float v = (col <= row) ? sc[t] : -1e30f;
            float r = v;
            r = fmaxf(r, __shfl_xor(r, 1, 32));
            r = fmaxf(r, __shfl_xor(r, 2, 32));
            r = fmaxf(r, __shfl_xor(r, 4, 32));
            r = fmaxf(r, __shfl_xor(r, 8, 32));
            const float mn = fmaxf(m_i[t], r);
            const float a  = __expf(m_i[t] - mn);
            const float pv = __expf(v - mn);
            float rs = pv;
            rs += __shfl_xor(rs, 1, 32);
            rs += __shfl_xor(rs, 2, 32);
            rs += __shfl_xor(rs, 4, 32);
            rs += __shfl_xor(rs, 8, 32);
            l_i[t] = l_i[t] * a + rs;
            m_i[t] = mn;
            alpha[t] = a;
            sc[t] = pv;
        }
        #pragma unroll
        for (int t = 0; t < 8; ++t) {
            o0[t] *= alpha[t]; o1[t] *= alpha[t];
            o2[t] *= alpha[t]; o3[t] *= alpha[t];
        }

        // Transpose P (C layout -> A layout) through LDS
        #pragma unroll
        for (int t = 0; t < 8; ++t) pbuf[t + 8 * half][l16] = sc[t];
        asm volatile("s_wait_dscnt 0" ::: "memory");
        v2f pa[4];
        #pragma unroll
        for (int kk = 0; kk < 4; ++kk) {
            pa[kk].x = pbuf[l16][kk * 4 + 2 * half];
            pa[kk].y = pbuf[l16][kk * 4 + 2 * half + 1];
        }

        // O += P @ V_tile  (4 col-tiles of 16, 4 k-steps)
        #pragma unroll
        for (int kk = 0; kk < 4; ++kk) {
            const int krow = kBase + kk * 4 + 2 * half;
            const float* v0 = Vp + (size_t)krow * DK_;
            const float* v1 = v0 + DK_;
            v2f b0; b0.x = v0[ 0 + l16]; b0.y = v1[ 0 + l16];
            v2f b1; b1.x = v0[16 + l16]; b1.y = v1[16 + l16];
            v2f b2; b2.x = v0[32 + l16]; b2.y = v1[32 + l16];
            v2f b3; b3.x = v0[48 + l16]; b3.y = v1[48 + l16];
            o0 = wmma_f32(pa[kk], b0, o0);
            o1 = wmma_f32(pa[kk], b1, o1);
            o2 = wmma_f32(pa[kk], b2, o2);
            o3 = wmma_f32(pa[kk], b3, o3);
        }
        asm volatile("s_wait_dscnt 0" ::: "memory"); // LDS reads done before next-iter stores
    }

    // normalize and write to (B,S,D)
    float* Orow = Oout + ((size_t)b * S_ + qBase) * D_ + h * DK_;
    #pragma unroll
    for (int t = 0; t < 8; ++t) {
        const float inv = 1.0f / l_i[t];
        float* dst = Orow + (size_t)(t + 8 * half) * D_;
        dst[ 0 + l16] = o0[t] * inv;
        dst[16 + l16] = o1[t] * inv;
        dst[32 + l16] = o2[t] * inv;
        dst[48 + l16] = o3[t] * inv;
    }
}

// ---------------------------------------------------------------------------
// Kernel 3: output projection. A:(4096,1024) @ Wo:(1024,1024) -> out.
// ---------------------------------------------------------------------------
__global__ __launch_bounds__(256) void out_proj_kernel(
    const float* __restrict__ A, const float* __restrict__ W,
    float* __restrict__ out) {
    const int lane = threadIdx.x & 31;
    const int wave = threadIdx.x >> 5;
    const int l16  = lane & 15;
    const int half = lane >> 4;
    const int rowBase = blockIdx.y * 32 + (wave >> 2) * 16;
    const int colBase = blockIdx.x * 64 + (wave & 3) * 16;

    const float* Arow = A + (size_t)(rowBase + l16) * D_ + 2 * half;
    const float* Bcol = W + (size_t)(2 * half) * D_ + colBase + l16;

    v8f acc = {};
    #pragma unroll 4
    for (int k = 0; k < D_; k += 4) {
        v2f a; a.x = Arow[k];              a.y = Arow[k + 1];
        v2f b; b.x = Bcol[(size_t)k * D_]; b.y = Bcol[(size_t)(k + 1) * D_];
        acc = wmma_f32(a, b, acc);
    }
    #pragma unroll
    for (int t = 0; t < 8; ++t) {
        const int gm = rowBase + t + 8 * half;
        const int gn = colBase + l16;
        out[(size_t)gm * D_ + gn] = acc[t];
    }
}

// ---------------------------------------------------------------------------
extern "C" void kernel_launch(void* const* d_in, const int* in_sizes, int n_in,
                              void* d_out, int out_size, void* d_ws, size_t ws_size,
                              hipStream_t stream) {
    (void)in_sizes; (void)n_in; (void)out_size; (void)ws_size;
    const float* X      = (const float*)d_in[0];   // (B,S,D) fp32
    const int*   tokpos = (const int*)d_in[1];     // (S) int32
    const float* Wqkv   = (const float*)d_in[2];   // (D,3D) fp32
    const float* Wo     = (const float*)d_in[3];   // (D,D) fp32
    float* out = (float*)d_out;

    float* ws = (float*)d_ws;
    const size_t BHSD = (size_t)B_ * H_ * S_ * DK_;   // 4,194,304 floats
    float* Qb = ws;
    float* Kb = ws + BHSD;
    float* Vb = ws + 2 * BHSD;
    float* Ob = ws + 3 * BHSD;                        // total 64 MB used

    dim3 g1(N3D_ / 64, MROWS_ / 32);                  // (48,128)
    qkv_rope_kernel<<<g1, 256, 0, stream>>>(X, Wqkv, tokpos, Qb, Kb, Vb);

    dim3 g2(S_ / 16, B_ * H_);                        // (128,32)
    flash_kernel<<<g2, 32, 0, stream>>>(Qb, Kb, Vb, Ob);

    dim3 g3(D_ / 64, MROWS_ / 32);                    // (16,128)
    out_proj_kernel<<<g3, 256, 0, stream>>>(Ob, Wo, out);
}